// PostProcessor_55946243997863
// MI455X (gfx1250) — compile-verified
//
#include <hip/hip_runtime.h>
#include <cstdint>

// ---------------------------------------------------------------------------
// Problem constants (from the reference)
// ---------------------------------------------------------------------------
#define NBATCH 2
#define NCLS   30
#define NHEAD  4
#define PNUM   20
#define NEGV   (-1e30f)
#define HW_A   13600   // level a: 100x136
#define HW_B   3400    // level b: 50x68
#define MAX_HW HW_A

struct RegPtrs { const float* r[8]; };   // [head*2 + level]

// ---------------------------------------------------------------------------
// CDNA5 Tensor Data Mover support (gfx1250). Guarded so host pass compiles.
// This toolchain exposes the 6-arg builtin:
//   (uint32x4 g0, int32x8 g1, int32x4 g2, int32x4 g3, int32x8 g4, i32 cpol)
// ---------------------------------------------------------------------------
#if defined(__gfx1250__) && __has_builtin(__builtin_amdgcn_tensor_load_to_lds)
#define USE_TDM 1
typedef unsigned int tdm_u32x4 __attribute__((ext_vector_type(4)));
typedef int          tdm_i32x8 __attribute__((ext_vector_type(8)));
typedef int          tdm_i32x4 __attribute__((ext_vector_type(4)));
#else
#define USE_TDM 0
#endif

// ---------------------------------------------------------------------------
// Shared det decode: dets[k] for anchor idx of (n, c).
// reg layout: (N, C*16, hw) -> reg[((n*480) + c*16 + k)*hw + idx]
// kx (k<8):  r*wa + cxa ; ky (k>=8): r*ha + cya
// ---------------------------------------------------------------------------
__device__ __forceinline__ void compute_dets(const float* __restrict__ reg,
                                             const float* __restrict__ anch,
                                             int hw, int n, int c, int idx,
                                             float* d) {
  float x1 = anch[idx * 4 + 0];
  float y1 = anch[idx * 4 + 1];
  float x2 = anch[idx * 4 + 2];
  float y2 = anch[idx * 4 + 3];
  float wa  = x2 - x1;
  float ha  = y2 - y1;
  float cxa = (x1 + x2) * 0.5f;
  float cya = (y1 + y2) * 0.5f;
  const float* base = reg + ((size_t)(n * (NCLS * 16) + c * 16)) * (size_t)hw + idx;
#pragma unroll
  for (int k = 0; k < 8; ++k)  d[k] = base[(size_t)k * hw] * wa + cxa;
#pragma unroll
  for (int k = 8; k < 16; ++k) d[k] = base[(size_t)k * hw] * ha + cya;
}

// ---------------------------------------------------------------------------
// Kernel A: per (n,c) row of one (head, level):
//   TDM-stage cls row into LDS, transform to sm = mask ? sqrt(sigmoid) : NEG,
//   count mask, then 20 exact argmax passes (ties -> lower index).
// Launched once per (head, level) with pre-offset output pointers.
// ---------------------------------------------------------------------------
__global__ __launch_bounds__(256) void topk_kernel(
    const float* __restrict__ cls, int hw,
    float* __restrict__ ts, int* __restrict__ ti, int* __restrict__ cnt) {
  // Single LDS block so the TDM tile lands at LDS offset 0.
  __shared__ __align__(16) float ldsf[MAX_HW + 512];
  float* smv   = ldsf;
  float* red_v = ldsf + MAX_HW;
  int*   red_i = (int*)(ldsf + MAX_HW + 256);

  const int tid = threadIdx.x;
  const int blk = blockIdx.x;                    // n*NCLS + c
  const float* src = cls + (size_t)blk * (size_t)hw;

#if USE_TDM
  if (tid < 32) {  // one wave issues the tensor DMA
    uint64_t ga = (uint64_t)(uintptr_t)src;
    tdm_u32x4 g0;
    g0[0] = 1u;                                            // count=1, is_restore=0
    g0[1] = 0u;                                            // lds_addr = 0 (ldsf base)
    g0[2] = (unsigned)ga;                                  // global_addr[31:0]
    g0[3] = ((unsigned)(ga >> 32) & 0x01FFFFFFu) |
            0x80000000u;                                   // addr[56:32] | type=2
    tdm_i32x8 g1;
    g1[0] = (int)(2u << 16);                               // data_size = 4 bytes
    g1[1] = (int)(((unsigned)hw & 0xFFFFu) << 16);         // tensor_dim0 lo16
    g1[2] = (int)(((unsigned)hw >> 16) | (1u << 16));      // tensor_dim0 hi16 | tensor_dim1=1
    g1[3] = (int)(((unsigned)hw & 0xFFFFu) << 16);         // tile_dim0 = hw
    g1[4] = 1;                                             // tile_dim1=1, tile_dim2=0
    g1[5] = hw;                                            // tensor_dim0_stride lo32
    g1[6] = 0;
    g1[7] = 0;
    tdm_i32x4 g2; g2[0] = 1; g2[1] = 1; g2[2] = 0; g2[3] = 0;   // dim2=dim3=1
    tdm_i32x4 g3; g3[0] = 0; g3[1] = (int)(1u << 16); g3[2] = 0; g3[3] = 0; // dim4=1
    tdm_i32x8 g4 = {0, 0, 0, 0, 0, 0, 0, 0};               // extra group (zero-filled)
    __builtin_amdgcn_tensor_load_to_lds(g0, g1, g2, g3, g4, 0);
  }
  __builtin_amdgcn_s_wait_tensorcnt(0);
  __syncthreads();
#else
  for (int i = tid; i < hw; i += 256) smv[i] = src[i];
  __syncthreads();
#endif

  // Transform in place + per-thread mask count
  int my_cnt = 0;
  for (int i = tid; i < hw; i += 256) {
    float x = smv[i];
    float s = 1.0f / (1.0f + expf(-x));
    bool  m = s > 0.1f;
    my_cnt += m ? 1 : 0;
    smv[i] = m ? sqrtf(s) : NEGV;
  }
  __syncthreads();

  // Count reduction
  red_i[tid] = my_cnt;
  __syncthreads();
  for (int srd = 128; srd > 0; srd >>= 1) {
    if (tid < srd) red_i[tid] += red_i[tid + srd];
    __syncthreads();
  }
  if (tid == 0) cnt[blk] = red_i[0];
  __syncthreads();

  // 20 exact argmax passes (jax.lax.top_k ordering: ties -> lower index)
  for (int k = 0; k < PNUM; ++k) {
    float bv  = -__builtin_inff();
    int   bix = 0x7FFFFFFF;
    for (int i = tid; i < hw; i += 256) {
      float v = smv[i];
      if (v > bv) { bv = v; bix = i; }     // ascending i -> first occurrence wins
    }
    red_v[tid] = bv;
    red_i[tid] = bix;
    __syncthreads();
    for (int srd = 128; srd > 0; srd >>= 1) {
      if (tid < srd) {
        float v2 = red_v[tid + srd];
        int   i2 = red_i[tid + srd];
        if (v2 > red_v[tid] || (v2 == red_v[tid] && i2 < red_i[tid])) {
          red_v[tid] = v2;
          red_i[tid] = i2;
        }
      }
      __syncthreads();
    }
    if (tid == 0) {
      ts[(size_t)blk * PNUM + k] = red_v[0];
      ti[(size_t)blk * PNUM + k] = red_i[0];
      smv[red_i[0]] = -__builtin_inff();   // knock out the winner
    }
    __syncthreads();
  }
}

// ---------------------------------------------------------------------------
// Kernel B: per (head, n, c) -> nk[2] soft-assigned proposal counts.
// ---------------------------------------------------------------------------
__global__ void nk_kernel(RegPtrs rp,
                          const float* __restrict__ anch_a,
                          const float* __restrict__ anch_b,
                          const float* __restrict__ ts,
                          const int*   __restrict__ ti,
                          int* __restrict__ nk_out) {
  int t = blockIdx.x * blockDim.x + threadIdx.x;
  if (t >= NHEAD * NBATCH * NCLS) return;
  int c = t % NCLS;
  int n = (t / NCLS) % NBATCH;
  int h = t / (NCLS * NBATCH);

  float best[2], szl[2];
  for (int l = 0; l < 2; ++l) {
    int p   = ((h * 2 + l) * NBATCH + n) * NCLS + c;
    float b = ts[p * PNUM + 0];            // top-1 == max over hw
    best[l] = b;
    float sz = 0.0f;
    if (b > 0.0f) {
      int idx = ti[p * PNUM + 0];          // top-1 index == argmax
      const float* anch = l ? anch_b : anch_a;
      int hw = l ? HW_B : HW_A;
      float d[16];
      compute_dets(rp.r[h * 2 + l], anch, hw, n, c, idx, d);
      float kxmx = d[0], kxmn = d[0], kymx = d[8], kymn = d[8];
#pragma unroll
      for (int k = 1; k < 8; ++k)  { kxmx = fmaxf(kxmx, d[k]); kxmn = fminf(kxmn, d[k]); }
#pragma unroll
      for (int k = 9; k < 16; ++k) { kymx = fmaxf(kymx, d[k]); kymn = fminf(kymn, d[k]); }
      sz = fmaxf(kxmx - kxmn, kymx - kymn);
    }
    szl[l] = (b > 0.0f) ? sz : 0.0f;
  }

  int   bi   = (best[1] > best[0]) ? 1 : 0;   // argmax over levels, ties -> 0
  float box  = szl[bi];
  float safe = (box > 0.0f) ? box : 1.0f;
  float dk0  = log2f(safe / 64.0f);
  float dk1  = log2f(safe / 128.0f);
  float e0   = expf(-dk0 * dk0);
  float e1   = expf(-dk1 * dk1);
  float ssum = e0 + e1;
  float nk0  = 20.0f * e0 / ssum;
  float nk1  = 20.0f * e1 / ssum;
  nk_out[t * 2 + 0] = (box > 0.0f) ? (int)(nk0 + 0.5f) : 0;
  nk_out[t * 2 + 1] = (box > 0.0f) ? (int)(nk1 + 0.5f) : 0;
}

// ---------------------------------------------------------------------------
// Kernel C: one thread per output row of 17 = (h, n, c, l, j).
// Output layout: (n, head, c, level, PNUM, 17) row-major.
// ---------------------------------------------------------------------------
__global__ __launch_bounds__(256) void emit_kernel(RegPtrs rp,
                                                   const float* __restrict__ anch_a,
                                                   const float* __restrict__ anch_b,
                                                   const float* __restrict__ ts,
                                                   const int*   __restrict__ ti,
                                                   const int*   __restrict__ cnt,
                                                   const int*   __restrict__ nk,
                                                   float* __restrict__ out) {
  int t = blockIdx.x * blockDim.x + threadIdx.x;
  const int TOT = NHEAD * NBATCH * NCLS * 2 * PNUM;
  if (t >= TOT) return;
  int j = t % PNUM;  int r = t / PNUM;
  int l = r % 2;     r /= 2;
  int c = r % NCLS;  r /= NCLS;
  int n = r % NBATCH;
  int h = r / NBATCH;

  int p     = ((h * 2 + l) * NBATCH + n) * NCLS + c;
  int nkIdx = ((h * NBATCH + n) * NCLS + c) * 2 + l;
  int cv    = cnt[p];
  int nv    = nk[nkIdx];
  int pk    = (cv < nv) ? cv : nv;

  float tsj = ts[p * PNUM + j];
  bool  vm  = (j < pk) && (tsj > 0.0f);

  size_t ob = ((((size_t)n * NHEAD + h) * NCLS + c) * 2 + l) * PNUM + j;
  float* o  = out + ob * 17;
  o[0] = vm ? tsj : 0.0f;
  if (vm) {
    int idx = ti[p * PNUM + j];
    const float* anch = l ? anch_b : anch_a;
    int hw = l ? HW_B : HW_A;
    float d[16];
    compute_dets(rp.r[h * 2 + l], anch, hw, n, c, idx, d);
#pragma unroll
    for (int k = 0; k < 16; ++k) o[1 + k] = d[k];
  } else {
#pragma unroll
    for (int k = 0; k < 16; ++k) o[1 + k] = 0.0f;
  }
}

// ---------------------------------------------------------------------------
// Launch
// Inputs: h=1..4 x {cls a, reg a, cls b, reg b} -> d_in[h*4 + l*2 (+1 for reg)]
//         anchors_a = d_in[16], anchors_b = d_in[17]
// ---------------------------------------------------------------------------
extern "C" void kernel_launch(void* const* d_in, const int* in_sizes, int n_in,
                              void* d_out, int out_size, void* d_ws, size_t ws_size,
                              hipStream_t stream) {
  (void)in_sizes; (void)n_in; (void)out_size; (void)ws_size;
  const int HWs[2]  = {HW_A, HW_B};
  const int NPROB   = NHEAD * 2 * NBATCH * NCLS;   // 480

  float* ws_ts  = (float*)d_ws;                    // 480*20 floats
  int*   ws_ti  = (int*)(ws_ts + NPROB * PNUM);    // 480*20 ints
  int*   ws_cnt = (int*)(ws_ti + NPROB * PNUM);    // 480 ints
  int*   ws_nk  = ws_cnt + NPROB;                  // 480 ints

  for (int h = 0; h < NHEAD; ++h) {
    for (int l = 0; l < 2; ++l) {
      const float* cls = (const float*)d_in[h * 4 + l * 2];
      int pbase = (h * 2 + l) * (NBATCH * NCLS);
      topk_kernel<<<NBATCH * NCLS, 256, 0, stream>>>(
          cls, HWs[l],
          ws_ts + (size_t)pbase * PNUM,
          ws_ti + (size_t)pbase * PNUM,
          ws_cnt + pbase);
    }
  }

  RegPtrs rp;
  for (int h = 0; h < NHEAD; ++h)
    for (int l = 0; l < 2; ++l)
      rp.r[h * 2 + l] = (const float*)d_in[h * 4 + l * 2 + 1];
  const float* anch_a = (const float*)d_in[16];
  const float* anch_b = (const float*)d_in[17];

  nk_kernel<<<(NHEAD * NBATCH * NCLS + 63) / 64, 64, 0, stream>>>(
      rp, anch_a, anch_b, ws_ts, ws_ti, ws_nk);

  emit_kernel<<<(NHEAD * NBATCH * NCLS * 2 * PNUM + 255) / 256, 256, 0, stream>>>(
      rp, anch_a, anch_b, ws_ts, ws_ti, ws_cnt, ws_nk, (float*)d_out);
}